// GPGkNN2_41188736368643
// MI455X (gfx1250) — compile-verified
//
#include <hip/hip_runtime.h>
#include <hip/hip_bf16.h>

// ---------------------------------------------------------------------------
// Problem constants (from the reference): B=2, N=8192, PHY=2, C=64, HID=64,
// FC=128, L=3, E=524288.  M = B*N = 16384 nodes.
// Roofline: edge gather/scatter dominates (~1 GB irregular traffic); GEMMs are
// 16384x64x64 fp32 -> use V_WMMA_F32_16X16X4_F32 (exact fp32 matrix path).
// ---------------------------------------------------------------------------
namespace {
constexpr int M_NODES = 16384;
constexpr int E_EDGES = 524288;
constexpr int C = 64;
constexpr int FC = 128;
constexpr float LN_EPS = 1e-5f;
constexpr float INV_PI = 0.31830988618379067f;
}

typedef float v2f __attribute__((ext_vector_type(2)));
typedef float v8f __attribute__((ext_vector_type(8)));

__device__ __forceinline__ float gelu_exact(float x) {
    return 0.5f * x * (1.0f + erff(x * 0.70710678118654752f));
}

// Reduce across the 16 lanes of each wave-half (masks 1,2,4,8 stay in-half).
__device__ __forceinline__ float half_reduce(float v) {
    v += __shfl_xor(v, 1, 32);
    v += __shfl_xor(v, 2, 32);
    v += __shfl_xor(v, 4, 32);
    v += __shfl_xor(v, 8, 32);
    return v;
}

// ---------------------------------------------------------------------------
// WMMA f32 16x16x4 helper: accumulate D += A_rows(16xK=64) * W^T into NT tiles
// of 16 columns each.  A is (M,64) row-major, W is (n_out,64) row-major so
// B[k][n] = W[n*64+k].
// A VGPR layout (32-bit A, 16x4): lanes 0-15 hold M=0..15 with K={0,1},
// lanes 16-31 hold M=0..15 with K={2,3}.  B mirrors this over N.
// ---------------------------------------------------------------------------
template <int NT>
__device__ __forceinline__ void wmma_gemm64(const float* __restrict__ A,
                                            const float* __restrict__ W,
                                            int m0, int lane, v8f acc[NT]) {
    const int half = lane >> 4;
    const int ln   = lane & 15;
    const int arow = m0 + ln;
#pragma unroll
    for (int kc = 0; kc < 16; ++kc) {
        const int k = kc * 4 + half * 2;
        v2f a;
        a.x = A[arow * 64 + k];
        a.y = A[arow * 64 + k + 1];
#pragma unroll
        for (int t = 0; t < NT; ++t) {
            const int n = t * 16 + ln;
            v2f b;
            b.x = W[n * 64 + k];
            b.y = W[n * 64 + k + 1];
            acc[t] = __builtin_amdgcn_wmma_f32_16x16x4_f32(
                false, a, false, b, (short)0, acc[t], false, false);
        }
    }
}

// ---------------------------------------------------------------------------
// h = LayerNorm(feat @ fc0_w^T + fc0_b): one wave per node, 2 channels/lane.
// ---------------------------------------------------------------------------
__global__ void node_init_kernel(const float* __restrict__ x,
                                 const float* __restrict__ w,   // (64,4)
                                 const float* __restrict__ b,
                                 const float* __restrict__ g,
                                 const float* __restrict__ be,
                                 float* __restrict__ h) {
    const int wid  = (blockIdx.x * blockDim.x + threadIdx.x) >> 5;
    const int lane = threadIdx.x & 31;
    if (wid >= M_NODES) return;
    const float* xr = x + wid * 5;
    const float f0 = xr[0], f1 = xr[1], f2 = xr[2], f3 = xr[3];
    float vals[2];
#pragma unroll
    for (int q = 0; q < 2; ++q) {
        const int c = lane + q * 32;
        vals[q] = f0 * w[c * 4 + 0] + f1 * w[c * 4 + 1] +
                  f2 * w[c * 4 + 2] + f3 * w[c * 4 + 3] + b[c];
    }
    float s  = vals[0] + vals[1];
    float ss = vals[0] * vals[0] + vals[1] * vals[1];
#pragma unroll
    for (int m = 1; m < 32; m <<= 1) {
        s  += __shfl_xor(s,  m, 32);
        ss += __shfl_xor(ss, m, 32);
    }
    const float mu  = s * (1.0f / 64.0f);
    const float var = fmaxf(ss * (1.0f / 64.0f) - mu * mu, 0.0f);
    const float rs  = rsqrtf(var + LN_EPS);
#pragma unroll
    for (int q = 0; q < 2; ++q) {
        const int c = lane + q * 32;
        h[wid * 64 + c] = (vals[q] - mu) * rs * g[c] + be[c];
    }
}

// ---------------------------------------------------------------------------
// Precompute per-edge (vec.x, vec.y, dist2, gw[src]) once; reused by 3 layers.
// ---------------------------------------------------------------------------
__global__ void edge_pre_kernel(const float* __restrict__ x,
                                const int* __restrict__ ei,
                                float4* __restrict__ edata) {
    const int e = blockIdx.x * blockDim.x + threadIdx.x;
    if (e >= E_EDGES) return;
    const int s = ei[e];
    const int d = ei[E_EDGES + e];
    const float vx = x[s * 5 + 2] - x[d * 5 + 2];
    const float vy = x[s * 5 + 3] - x[d * 5 + 3];
    edata[e] = make_float4(vx, vy, vx * vx + vy * vy, x[s * 5 + 4]);
}

__global__ void zero_kernel(float* __restrict__ p, int n) {
    const int i = blockIdx.x * blockDim.x + threadIdx.x;
    if (i < n) p[i] = 0.0f;
}

// ---------------------------------------------------------------------------
// hl = h @ gg_fc1_w[l]^T + b : one wave per 16 rows, 4 N-tiles of WMMA.
// ---------------------------------------------------------------------------
__global__ void gemm_hl_kernel(const float* __restrict__ A,
                               const float* __restrict__ W,
                               const float* __restrict__ bias,
                               float* __restrict__ out) {
    const int wid  = (blockIdx.x * blockDim.x + threadIdx.x) >> 5;
    const int lane = threadIdx.x & 31;
    const int m0   = wid * 16;
    if (m0 >= M_NODES) return;
    const v8f z = {0.f, 0.f, 0.f, 0.f, 0.f, 0.f, 0.f, 0.f};
    v8f acc[4] = {z, z, z, z};
    wmma_gemm64<4>(A, W, m0, lane, acc);
    const int half = lane >> 4;
    const int ln   = lane & 15;
#pragma unroll
    for (int t = 0; t < 4; ++t) {
        const int   n  = t * 16 + ln;
        const float bn = bias[n];
#pragma unroll
        for (int v = 0; v < 8; ++v) {
            const int row = m0 + v + 8 * half;
            out[row * 64 + n] = acc[t][v] + bn;
        }
    }
}

// ---------------------------------------------------------------------------
// Heavy edge kernel: 16 threads per edge, 4 HID channels each.
// val = (w/pi)*exp(-w*dist2) * sin(vec . freq) * hl[src] * gw[src]
// scatter-add into agg[dst] with hardware f32 atomics.
// ---------------------------------------------------------------------------
__global__ void edge_main_kernel(const float4* __restrict__ edata,
                                 const int* __restrict__ ei,
                                 const float* __restrict__ hl,
                                 const float* __restrict__ wvec,  // (64)
                                 const float* __restrict__ freq,  // (2,64)
                                 float* __restrict__ agg) {
    const int tid = blockIdx.x * blockDim.x + threadIdx.x;
    const int e   = tid >> 4;
    if (e >= E_EDGES) return;
    const int j = tid & 15;
    const float4 ed = edata[e];
    const int s = ei[e];
    const int d = ei[E_EDGES + e];
    const float4 h4 = *(const float4*)(hl + s * 64 + j * 4);
    const float hv[4] = {h4.x, h4.y, h4.z, h4.w};
    float* ap = agg + d * 64 + j * 4;
#pragma unroll
    for (int q = 0; q < 4; ++q) {
        const int   hid   = j * 4 + q;
        const float w     = wvec[hid];
        const float gauss = (w * INV_PI) * __expf(-w * ed.z);     // PHY==2
        const float four  = __sinf(ed.x * freq[hid] + ed.y * freq[64 + hid]);
        unsafeAtomicAdd(ap + q, gauss * four * hv[q] * ed.w);
    }
}

// ---------------------------------------------------------------------------
// h = LN(agg@W2^T + b2 + h@Wc^T + bc) [+ GELU].  In-place on h (each wave
// reads/writes only its own 16 rows).  Row stats via in-half shfl reductions:
// C-layout puts row M=v+8*half entirely in the 16 lanes of one half.
// ---------------------------------------------------------------------------
__global__ void gemm_dual_ln_kernel(const float* __restrict__ A1,
                                    const float* __restrict__ W1,
                                    const float* __restrict__ b1,
                                    const float* __restrict__ A2,
                                    const float* __restrict__ W2,
                                    const float* __restrict__ b2,
                                    const float* __restrict__ g,
                                    const float* __restrict__ be,
                                    float* __restrict__ out,
                                    int apply_gelu) {
    const int wid  = (blockIdx.x * blockDim.x + threadIdx.x) >> 5;
    const int lane = threadIdx.x & 31;
    const int m0   = wid * 16;
    if (m0 >= M_NODES) return;
    const v8f z = {0.f, 0.f, 0.f, 0.f, 0.f, 0.f, 0.f, 0.f};
    v8f acc[4] = {z, z, z, z};
    wmma_gemm64<4>(A1, W1, m0, lane, acc);   // x1 = agg @ fc2^T
    wmma_gemm64<4>(A2, W2, m0, lane, acc);   // x2 = h @ wc^T (accumulated)
    const int half = lane >> 4;
    const int ln   = lane & 15;
#pragma unroll
    for (int t = 0; t < 4; ++t) {
        const int   n  = t * 16 + ln;
        const float bn = b1[n] + b2[n];
#pragma unroll
        for (int v = 0; v < 8; ++v) acc[t][v] += bn;
    }
#pragma unroll
    for (int v = 0; v < 8; ++v) {
        float s  = acc[0][v] + acc[1][v] + acc[2][v] + acc[3][v];
        float ss = acc[0][v] * acc[0][v] + acc[1][v] * acc[1][v] +
                   acc[2][v] * acc[2][v] + acc[3][v] * acc[3][v];
        s  = half_reduce(s);
        ss = half_reduce(ss);
        const float mu  = s * (1.0f / 64.0f);
        const float var = fmaxf(ss * (1.0f / 64.0f) - mu * mu, 0.0f);
        const float rs  = rsqrtf(var + LN_EPS);
        const int   row = m0 + v + 8 * half;
#pragma unroll
        for (int t = 0; t < 4; ++t) {
            const int n = t * 16 + ln;
            float y = (acc[t][v] - mu) * rs * g[n] + be[n];
            if (apply_gelu) y = gelu_exact(y);
            out[row * 64 + n] = y;
        }
    }
}

// ---------------------------------------------------------------------------
// out[m] = gelu(h[m,:]@fc1^T + b1) . fc2_w + fc2_b : 8 WMMA N-tiles then a
// cross-lane dot reduction; one scalar per row.
// ---------------------------------------------------------------------------
__global__ void final_kernel(const float* __restrict__ A,
                             const float* __restrict__ W1,   // (128,64)
                             const float* __restrict__ b1,   // (128)
                             const float* __restrict__ w2,   // (128)
                             const float* __restrict__ b2,   // (1)
                             float* __restrict__ out) {
    const int wid  = (blockIdx.x * blockDim.x + threadIdx.x) >> 5;
    const int lane = threadIdx.x & 31;
    const int m0   = wid * 16;
    if (m0 >= M_NODES) return;
    const v8f z = {0.f, 0.f, 0.f, 0.f, 0.f, 0.f, 0.f, 0.f};
    v8f acc[8] = {z, z, z, z, z, z, z, z};
    wmma_gemm64<8>(A, W1, m0, lane, acc);
    const int half = lane >> 4;
    const int ln   = lane & 15;
    float r[8];
#pragma unroll
    for (int v = 0; v < 8; ++v) r[v] = 0.0f;
#pragma unroll
    for (int t = 0; t < 8; ++t) {
        const int   n  = t * 16 + ln;
        const float bn = b1[n];
        const float wn = w2[n];
#pragma unroll
        for (int v = 0; v < 8; ++v)
            r[v] += gelu_exact(acc[t][v] + bn) * wn;
    }
#pragma unroll
    for (int v = 0; v < 8; ++v) r[v] = half_reduce(r[v]);
    if (ln == 0) {
        const float bias = b2[0];
#pragma unroll
        for (int v = 0; v < 8; ++v)
            out[m0 + v + 8 * half] = r[v] + bias;
    }
}

// ---------------------------------------------------------------------------
extern "C" void kernel_launch(void* const* d_in, const int* in_sizes, int n_in,
                              void* d_out, int out_size, void* d_ws, size_t ws_size,
                              hipStream_t stream) {
    const float* x        = (const float*)d_in[0];
    const int*   ei       = (const int*)d_in[1];
    const float* fc0_w    = (const float*)d_in[2];
    const float* fc0_b    = (const float*)d_in[3];
    const float* ln0_g    = (const float*)d_in[4];
    const float* ln0_b    = (const float*)d_in[5];
    const float* gg_fc1_w = (const float*)d_in[6];
    const float* gg_fc1_b = (const float*)d_in[7];
    const float* gg_fc2_w = (const float*)d_in[8];
    const float* gg_fc2_b = (const float*)d_in[9];
    const float* gg_wt    = (const float*)d_in[10];
    const float* gg_freq  = (const float*)d_in[11];
    const float* wc_w     = (const float*)d_in[12];
    const float* wc_b     = (const float*)d_in[13];
    const float* ln_g     = (const float*)d_in[14];
    const float* ln_b     = (const float*)d_in[15];
    const float* fc1_w    = (const float*)d_in[16];
    const float* fc1_b    = (const float*)d_in[17];
    const float* fc2_w    = (const float*)d_in[18];
    const float* fc2_b    = (const float*)d_in[19];
    (void)in_sizes; (void)n_in; (void)out_size; (void)ws_size;

    // Workspace layout (floats): edata[E*4] | h[M*64] | hl[M*64] | agg[M*64]
    float*  ws    = (float*)d_ws;
    float4* edata = (float4*)ws;
    float*  h     = ws + (size_t)E_EDGES * 4;
    float*  hl    = h + (size_t)M_NODES * 64;
    float*  agg   = hl + (size_t)M_NODES * 64;

    const int TPB = 256;
    // 1) node features: fc0 + LN  (one wave per node -> 8 nodes/block)
    node_init_kernel<<<M_NODES / 8, TPB, 0, stream>>>(x, fc0_w, fc0_b, ln0_g, ln0_b, h);
    // 2) per-edge geometry precompute
    edge_pre_kernel<<<(E_EDGES + TPB - 1) / TPB, TPB, 0, stream>>>(x, ei, edata);

    const int gemm_blocks = (M_NODES / 16) / 8;             // 1024 waves / 8 per block
    const int edge_blocks = (E_EDGES * 16) / TPB;           // 16 threads per edge
    const int zero_blocks = (M_NODES * 64 + TPB - 1) / TPB;

    for (int l = 0; l < 3; ++l) {
        gemm_hl_kernel<<<gemm_blocks, TPB, 0, stream>>>(
            h, gg_fc1_w + l * 4096, gg_fc1_b + l * 64, hl);
        zero_kernel<<<zero_blocks, TPB, 0, stream>>>(agg, M_NODES * 64);
        edge_main_kernel<<<edge_blocks, TPB, 0, stream>>>(
            edata, ei, hl, gg_wt + l * 64, gg_freq + l * 128, agg);
        gemm_dual_ln_kernel<<<gemm_blocks, TPB, 0, stream>>>(
            agg, gg_fc2_w + l * 4096, gg_fc2_b + l * 64,
            h,   wc_w     + l * 4096, wc_b     + l * 64,
            ln_g + l * 64, ln_b + l * 64, h, (l < 2) ? 1 : 0);
    }
    // 3) head: gelu(h@fc1^T+b) @ fc2^T + b
    final_kernel<<<gemm_blocks, TPB, 0, stream>>>(h, fc1_w, fc1_b, fc2_w, fc2_b,
                                                  (float*)d_out);
}